// SSIMLoss_8143257993924
// MI455X (gfx1250) — compile-verified
//
#include <hip/hip_runtime.h>

typedef __attribute__((ext_vector_type(2))) float v2f;
typedef __attribute__((ext_vector_type(8))) float v8f;

#define BATCH 32
#define CHN   3
#define HH    256
#define WW    800
#define KH    16
#define KW    20
#define SH    8
#define SW    10
#define PH    31          // (256-16)/8 + 1
#define PW    79          // (800-20)/10 + 1
#define JPAD  80          // padded column count (16-aligned tiles: 5 tiles)
#define PLANE (HH*JPAD)   // 20480 floats per (b,c,q) plane
#define NPATCH (BATCH*PH*PW)

// ---------------------------------------------------------------------------
// Stage 1: per-(b,c,h) row, compute 5 column-window sums (KW=20, SW=10):
//   q0=sum(x1), q1=sum(x2), q2=sum(x1^2), q3=sum(x2^2), q4=sum(x1*x2)
// Row loaded once into LDS with b128 loads; each input byte read exactly once.
// Output layout: ws[((b*3+c)*5+q)*PLANE + h*JPAD + j], j=79 column zero-padded.
// ---------------------------------------------------------------------------
__global__ void ssim_colsum_kernel(const float* __restrict__ x1,
                                   const float* __restrict__ x2,
                                   float* __restrict__ ws) {
  const int rid = blockIdx.x;        // [0, 32*3*256)
  const int tid = threadIdx.x;       // 128 threads
  const int bc  = rid >> 8;          // b*3 + c
  const int h   = rid & 255;

  __shared__ __align__(16) float r1[WW];
  __shared__ __align__(16) float r2[WW];
  const float4* g1 = (const float4*)(x1 + (size_t)rid * WW);
  const float4* g2 = (const float4*)(x2 + (size_t)rid * WW);
  for (int i = tid; i < WW / 4; i += 128) {
    ((float4*)r1)[i] = g1[i];
    ((float4*)r2)[i] = g2[i];
  }
  __syncthreads();

  if (tid < JPAD) {
    float o0 = 0.f, o1 = 0.f, o2 = 0.f, o3 = 0.f, o4 = 0.f;
    if (tid < PW) {
      const int base = tid * SW;
#pragma unroll
      for (int w = 0; w < KW; ++w) {
        float a = r1[base + w];
        float b = r2[base + w];
        o0 += a; o1 += b; o2 += a * a; o3 += b * b; o4 += a * b;
      }
    }
    float* p = ws + (size_t)bc * (5 * PLANE) + (size_t)h * JPAD + tid;
    p[0 * PLANE] = o0;
    p[1 * PLANE] = o1;
    p[2 * PLANE] = o2;
    p[3 * PLANE] = o3;
    p[4 * PLANE] = o4;
  }
}

// ---------------------------------------------------------------------------
// Stage 2: H-direction banded reduction as f32 WMMA matmul.
//   P[i][j] = sum_h M[i][h] * colsum[h][j],  M[i][h]=1 iff i*8 <= h < i*8+16
// One wave (32 threads) per (b, row-tile t in {0,1}, 16-wide j-tile).
// A (16x4 f32, band mask), B (4x16 f32 from colsum plane), C/D f32 16x16,
// via V_WMMA_F32_16X16X4_F32 — full f32 precision. 5 accumulators (one per
// statistic), then SSIM per element, wave reduction, atomic mean.
// ---------------------------------------------------------------------------
__global__ void ssim_wmma_kernel(const float* __restrict__ ws,
                                 float* __restrict__ out) {
  const int id    = blockIdx.x;      // [0, 32*2*5)
  const int lane  = threadIdx.x;     // 32 threads = 1 wave (wave32)
  const int b     = id / 10;
  const int rem   = id % 10;
  const int t     = rem / 5;         // patch-row tile: rows t*16 .. t*16+15
  const int jtile = rem % 5;

  const int ncol = jtile * 16 + (lane & 15);   // B/D column (N)
  const int am   = lane & 15;                  // A row (M)
  const int koff = (lane >> 4) << 1;           // K offset within chunk: 0 or 2
  const int lo   = t * 128 + am * 8;           // band start for A row am
  const int hbeg = t ? 128 : 0;
  const int hend = t ? 256 : 136;              // rows needed by this tile

  v8f acc[5];
#pragma unroll
  for (int q = 0; q < 5; ++q) acc[q] = (v8f){0.f, 0.f, 0.f, 0.f, 0.f, 0.f, 0.f, 0.f};

  for (int c = 0; c < CHN; ++c) {
    const float* plane0 = ws + (size_t)((b * CHN + c) * 5) * PLANE;
    for (int h0 = hbeg; h0 < hend; h0 += 4) {
      const int hk = h0 + koff;
      v2f a;
      a.x = (hk     >= lo && hk     < lo + KH) ? 1.0f : 0.0f;
      a.y = (hk + 1 >= lo && hk + 1 < lo + KH) ? 1.0f : 0.0f;
      const float* bp = plane0 + (size_t)hk * JPAD + ncol;
#pragma unroll
      for (int q = 0; q < 5; ++q) {
        v2f bb;
        bb.x = bp[q * PLANE];          // row hk
        bb.y = bp[q * PLANE + JPAD];   // row hk+1
        acc[q] = __builtin_amdgcn_wmma_f32_16x16x4_f32(
            /*neg_a=*/false, a, /*neg_b=*/false, bb,
            /*c_mod=*/(short)0, acc[q], /*reuse_a=*/false, /*reuse_b=*/false);
      }
    }
  }

  // SSIM per (patch-row, patch-col) element held by this lane.
  const float inv_n   = 1.0f / 960.0f;   // n = C*KH*KW
  const float inv_nm1 = 1.0f / 959.0f;
  const float c1 = 1e-8f, c2 = 1e-8f;
  const int   mb = (lane >> 4) * 8;      // C/D: lanes 16-31 hold M = r+8
  float lsum = 0.0f;
#pragma unroll
  for (int r = 0; r < 8; ++r) {
    float s1  = acc[0][r], s2  = acc[1][r];
    float s11 = acc[2][r], s22 = acc[3][r], s12 = acc[4][r];
    int   i   = t * 16 + mb + r;         // global patch row
    float mu1 = s1 * inv_n, mu2 = s2 * inv_n;
    float var1 = (s11 - s1 * mu1) * inv_nm1;
    float var2 = (s22 - s2 * mu2) * inv_nm1;
    float cov  = (s12 - s1 * mu2) * inv_nm1;
    float num = (2.0f * mu1 * mu2 + c1) * (2.0f * cov + c2);
    float den = (mu1 * mu1 + mu2 * mu2 + c1) * (var1 + var2 + c2);
    if (i < PH && ncol < PW) lsum += num / den;
  }

  // wave32 reduction, then atomic accumulate of the mean.
  for (int off = 16; off > 0; off >>= 1)
    lsum += __shfl_xor(lsum, off, 32);
  if (lane == 0)
    atomicAdd(out, lsum * (1.0f / (float)NPATCH));
}

// ---------------------------------------------------------------------------
extern "C" void kernel_launch(void* const* d_in, const int* in_sizes, int n_in,
                              void* d_out, int out_size, void* d_ws, size_t ws_size,
                              hipStream_t stream) {
  const float* x1 = (const float*)d_in[0];
  const float* x2 = (const float*)d_in[1];
  float* out = (float*)d_out;
  float* ws  = (float*)d_ws;   // needs 32*3*5*PLANE*4 = ~39.3 MB

  // Stage 1: column-window sums (reads each input element exactly once).
  ssim_colsum_kernel<<<BATCH * CHN * HH, 128, 0, stream>>>(x1, x2, ws);

  // Zero the scalar output (graph-capture-safe), then WMMA reduction stage.
  hipMemsetAsync(out, 0, sizeof(float), stream);
  ssim_wmma_kernel<<<BATCH * 2 * 5, 32, 0, stream>>>(ws, out);
}